// GATv2_50680614093209
// MI455X (gfx1250) — compile-verified
//
#include <hip/hip_runtime.h>
#include <hip/hip_bf16.h>

#define IN_DIM 64
#define HHID   128   // H * HID
#define HEADS  4
#define OUTC   32

typedef float v2f __attribute__((ext_vector_type(2)));
typedef float v8f __attribute__((ext_vector_type(8)));

__device__ __forceinline__ float wred_sum(float v) {
#pragma unroll
  for (int o = 16; o > 0; o >>= 1) v += __shfl_xor(v, o, 32);
  return v;
}
__device__ __forceinline__ float wred_max(float v) {
#pragma unroll
  for (int o = 16; o > 0; o >>= 1) v = fmaxf(v, __shfl_xor(v, o, 32));
  return v;
}

// ---------------------- CSR construction (by dst, incl. self loops) ----------------------
__global__ void k_zero_int(int* __restrict__ p, int n) {
  int i = blockIdx.x * blockDim.x + threadIdx.x;
  if (i < n) p[i] = 0;
}

__global__ void k_count(const int* __restrict__ ei, int* __restrict__ cnt, int E, int N) {
  int e = blockIdx.x * blockDim.x + threadIdx.x;
  if (e >= E + N) return;
  int d = (e < E) ? ei[E + e] : (e - E);   // row 1 of edge_index = dst; tail = self loops
  atomicAdd(&cnt[d], 1);
}

__global__ __launch_bounds__(1024) void k_exscan(const int* __restrict__ cnt,
                                                 int* __restrict__ rowptr, int n) {
  __shared__ int wsum[32];
  __shared__ int running;
  int lane = threadIdx.x & 31, wid = threadIdx.x >> 5;
  int nw = blockDim.x >> 5;
  if (threadIdx.x == 0) running = 0;
  __syncthreads();
  for (int base = 0; base < n; base += blockDim.x) {
    int i = base + threadIdx.x;
    int v = (i < n) ? cnt[i] : 0;
    int x = v;
#pragma unroll
    for (int o = 1; o < 32; o <<= 1) { int t = __shfl_up(x, o, 32); if (lane >= o) x += t; }
    if (lane == 31) wsum[wid] = x;
    __syncthreads();
    if (wid == 0) {
      int w = (lane < nw) ? wsum[lane] : 0;
#pragma unroll
      for (int o = 1; o < 32; o <<= 1) { int t = __shfl_up(w, o, 32); if (lane >= o) w += t; }
      wsum[lane] = w;   // inclusive scan of warp totals
    }
    __syncthreads();
    int woff = wid ? wsum[wid - 1] : 0;
    if (i < n) rowptr[i] = running + woff + (x - v);
    __syncthreads();
    if (threadIdx.x == 0) running += wsum[nw - 1];
    __syncthreads();
  }
  if (threadIdx.x == 0) rowptr[n] = running;
}

__global__ void k_copy_int(const int* __restrict__ a, int* __restrict__ b, int n) {
  int i = blockIdx.x * blockDim.x + threadIdx.x;
  if (i < n) b[i] = a[i];
}

__global__ void k_fill(const int* __restrict__ ei, int* __restrict__ cursor,
                       int* __restrict__ col, int E, int N) {
  int e = blockIdx.x * blockDim.x + threadIdx.x;
  if (e >= E + N) return;
  int d = (e < E) ? ei[E + e] : (e - E);
  int pos = atomicAdd(&cursor[d], 1);
  col[pos] = e;   // store edge id
}

// sort each row's edge ids -> deterministic accumulation order
__global__ void k_sortrows(const int* __restrict__ rowptr, int* __restrict__ col, int N) {
  int n = blockIdx.x * blockDim.x + threadIdx.x;
  if (n >= N) return;
  int r0 = rowptr[n], r1 = rowptr[n + 1];
  for (int i = r0 + 1; i < r1; ++i) {
    int key = col[i];
    int j = i - 1;
    while (j >= r0 && col[j] > key) { col[j + 1] = col[j]; --j; }
    col[j + 1] = key;
  }
}

// ---------------------- fp32 WMMA GEMM: Out[N,Dout] = X[N,Din] @ W[Din,Dout] ----------------------
__global__ __launch_bounds__(128) void k_gemm_wmma(const float* __restrict__ X,
                                                   const float* __restrict__ W,
                                                   float* __restrict__ Out,
                                                   int Nrows, int Din, int Dout) {
  int lane = threadIdx.x & 31;
  int wid  = threadIdx.x >> 5;                 // 4 waves per block, one 16x16 tile each
  int tilesM = (Nrows + 15) >> 4;
  int tm = blockIdx.x * 4 + wid;
  if (tm >= tilesM) return;                    // wave-uniform: EXEC stays all-ones
  int m0 = tm << 4;
  int n0 = blockIdx.y << 4;
  int r  = lane & 15;                          // M (for A) / N (for B,C)
  int kk = (lane >> 4) << 1;                   // K sub-pair: lanes 0-15 -> K0,K1 ; 16-31 -> K2,K3
  int mrow = m0 + r; if (mrow > Nrows - 1) mrow = Nrows - 1;
  const float* Ap = X + (size_t)mrow * Din + kk;
  const float* Bp = W + (size_t)kk * Dout + n0 + r;
  v8f c = {0.f, 0.f, 0.f, 0.f, 0.f, 0.f, 0.f, 0.f};
  for (int k = 0; k < Din; k += 4) {
    v2f a = *(const v2f*)(Ap + k);             // A[M][kk+k], A[M][kk+k+1]  (8B aligned)
    v2f b;
    b.x = Bp[(size_t)k * Dout];                // B[kk+k][N]
    b.y = Bp[(size_t)(k + 1) * Dout];          // B[kk+k+1][N]
    c = __builtin_amdgcn_wmma_f32_16x16x4_f32(false, a, false, b, (short)0, c, false, false);
  }
  int rbase = m0 + ((lane >> 4) << 3);         // VGPR v -> row M = v (+8 for upper half-wave)
  float* Op = Out + (size_t)rbase * Dout + n0 + r;
#pragma unroll
  for (int v = 0; v < 8; ++v)
    if (rbase + v < Nrows) Op[(size_t)v * Dout] = c[v];
}

// ---------------------- attention scalars: a_s[n,h], a_d[n,h] ----------------------
__global__ void k_attdots(const float* __restrict__ hin, const float* __restrict__ atts,
                          const float* __restrict__ attd, float* __restrict__ as_out,
                          float* __restrict__ ad_out, int N, int HH) {
  int idx = blockIdx.x * blockDim.x + threadIdx.x;   // idx = n*HH + h
  if (idx >= N * HH) return;
  int h = idx % HH;
  const float4* hp = (const float4*)(hin + (size_t)idx * 32);  // row (n, h*32..h*32+31)
  const float4* sp = (const float4*)(atts + h * 32);
  const float4* dp = (const float4*)(attd + h * 32);
  float s = 0.f, d = 0.f;
#pragma unroll
  for (int i = 0; i < 8; ++i) {
    float4 hv = hp[i], sv = sp[i], dv = dp[i];
    s += hv.x * sv.x + hv.y * sv.y + hv.z * sv.z + hv.w * sv.w;
    d += hv.x * dv.x + hv.y * dv.y + hv.z * dv.z + hv.w * dv.w;
  }
  as_out[idx] = s;
  ad_out[idx] = d;
}

// ---------------------- per-node softmax + gather + bias (+LN+ELU) ----------------------
// one wave32 per node; lane = channel within head
template <int HH, bool DO_LN>
__global__ __launch_bounds__(256) void k_aggregate(
    const float* __restrict__ hin, const float* __restrict__ a_s,
    const float* __restrict__ a_d, const int* __restrict__ rowptr,
    const int* __restrict__ col, const int* __restrict__ ei,
    const float* __restrict__ bias, const float* __restrict__ gamma,
    const float* __restrict__ beta, float* __restrict__ out, int N, int E) {
  const int stride = HH * 32;
  int lane = threadIdx.x & 31;
  int wid  = threadIdx.x >> 5;
  int n = blockIdx.x * 8 + wid;
  if (n >= N) return;
  int r0 = rowptr[n], r1 = rowptr[n + 1];
  float accs[HH];
#pragma unroll
  for (int h = 0; h < HH; ++h) {
    float adn = a_d[n * HH + h];
    // pass 1: segment max (lanes stride edges)
    float ml = -3.402823466e38f;
    for (int e = r0 + lane; e < r1; e += 32) {
      int eid = col[e];
      int s = (eid < E) ? ei[eid] : (eid - E);
      float tv = a_s[s * HH + h] + adn;
      tv = tv > 0.f ? tv : 0.2f * tv;
      ml = fmaxf(ml, tv);
    }
    float m = wred_max(ml);
    // pass 2: sum of exp
    float sl = 0.f;
    for (int e = r0 + lane; e < r1; e += 32) {
      int eid = col[e];
      int s = (eid < E) ? ei[eid] : (eid - E);
      float tv = a_s[s * HH + h] + adn;
      tv = tv > 0.f ? tv : 0.2f * tv;
      sl += __expf(tv - m);
    }
    float inv = 1.f / (wred_sum(sl) + 1e-16f);
    // pass 3: alpha-weighted gather of h[src] (all lanes same edge, coalesced 128B row)
    float acc = 0.f;
    for (int e = r0; e < r1; ++e) {
      int eid = col[e];
      int s = (eid < E) ? ei[eid] : (eid - E);
      if (e + 1 < r1) {
        int e2 = col[e + 1];
        int s2 = (e2 < E) ? ei[e2] : (e2 - E);
        __builtin_prefetch(hin + (size_t)s2 * stride + h * 32 + lane, 0, 1);
      }
      float tv = a_s[s * HH + h] + adn;
      tv = tv > 0.f ? tv : 0.2f * tv;
      float alpha = __expf(tv - m) * inv;
      acc = fmaf(alpha, hin[(size_t)s * stride + h * 32 + lane], acc);
    }
    accs[h] = acc + bias[h * 32 + lane];
  }
  if (DO_LN) {
    // LayerNorm across all HH*32 channels held by this wave, then ELU
    float p = 0.f;
#pragma unroll
    for (int h = 0; h < HH; ++h) p += accs[h];
    float mu = wred_sum(p) * (1.f / (float)stride);
    float vp = 0.f;
#pragma unroll
    for (int h = 0; h < HH; ++h) { float dd = accs[h] - mu; vp += dd * dd; }
    float var = wred_sum(vp) * (1.f / (float)stride);
    float rinv = rsqrtf(var + 1e-5f);
#pragma unroll
    for (int h = 0; h < HH; ++h) {
      float y = gamma[h * 32 + lane] * (accs[h] - mu) * rinv + beta[h * 32 + lane];
      y = y > 0.f ? y : (__expf(y) - 1.f);   // ELU(alpha=1)
      out[(size_t)n * stride + h * 32 + lane] = y;
    }
  } else {
#pragma unroll
    for (int h = 0; h < HH; ++h)
      out[(size_t)n * stride + h * 32 + lane] = accs[h];
  }
}

// ---------------------- launcher ----------------------
extern "C" void kernel_launch(void* const* d_in, const int* in_sizes, int n_in,
                              void* d_out, int out_size, void* d_ws, size_t ws_size,
                              hipStream_t stream) {
  const float* x0 = (const float*)d_in[0];
  const int* ei   = (const int*)d_in[1];
  int N  = in_sizes[0] / IN_DIM;
  int E  = in_sizes[1] / 2;
  int EL = E + N;

  const float* Wm[4]  = {(const float*)d_in[2],  (const float*)d_in[8],
                         (const float*)d_in[14], (const float*)d_in[20]};
  const float* ASm[4] = {(const float*)d_in[3],  (const float*)d_in[9],
                         (const float*)d_in[15], (const float*)d_in[21]};
  const float* ADm[4] = {(const float*)d_in[4],  (const float*)d_in[10],
                         (const float*)d_in[16], (const float*)d_in[22]};
  const float* Bm[4]  = {(const float*)d_in[5],  (const float*)d_in[11],
                         (const float*)d_in[17], (const float*)d_in[23]};
  const float* Gm[3]  = {(const float*)d_in[6],  (const float*)d_in[12], (const float*)d_in[18]};
  const float* BEm[3] = {(const float*)d_in[7],  (const float*)d_in[13], (const float*)d_in[19]};

  char* base = (char*)d_ws;
  size_t off = 0;
  auto alloc = [&](size_t bytes) {
    char* p = base + off;
    off += (bytes + 255) & ~(size_t)255;
    return p;
  };
  float* hbuf  = (float*)alloc((size_t)N * HHID * sizeof(float));
  float* xbuf  = (float*)alloc((size_t)N * HHID * sizeof(float));
  float* asb   = (float*)alloc((size_t)N * HEADS * sizeof(float));
  float* adb   = (float*)alloc((size_t)N * HEADS * sizeof(float));
  int* rowptr  = (int*)alloc((size_t)(N + 1) * sizeof(int));
  int* cnt     = (int*)alloc((size_t)N * sizeof(int));
  int* col     = (int*)alloc((size_t)EL * sizeof(int));
  (void)ws_size; (void)n_in; (void)out_size;

  const int T = 256;
  // CSR by dst (rebuilt each call; fully overwrites its buffers -> replay-safe)
  k_zero_int<<<(N + T - 1) / T, T, 0, stream>>>(cnt, N);
  k_count<<<(EL + T - 1) / T, T, 0, stream>>>(ei, cnt, E, N);
  k_exscan<<<1, 1024, 0, stream>>>(cnt, rowptr, N);
  k_copy_int<<<(N + T - 1) / T, T, 0, stream>>>(rowptr, cnt, N);
  k_fill<<<(EL + T - 1) / T, T, 0, stream>>>(ei, cnt, col, E, N);
  k_sortrows<<<(N + T - 1) / T, T, 0, stream>>>(rowptr, col, N);

  int tilesM = (N + 15) / 16;
  dim3 ggrid((tilesM + 3) / 4, HHID / 16);
  dim3 agrid((N + 7) / 8);

  // layer 0: 64 -> 128, GAT + LN + ELU
  k_gemm_wmma<<<ggrid, 128, 0, stream>>>(x0, Wm[0], hbuf, N, IN_DIM, HHID);
  k_attdots<<<(N * HEADS + T - 1) / T, T, 0, stream>>>(hbuf, ASm[0], ADm[0], asb, adb, N, HEADS);
  k_aggregate<HEADS, true><<<agrid, 256, 0, stream>>>(hbuf, asb, adb, rowptr, col, ei,
                                                      Bm[0], Gm[0], BEm[0], xbuf, N, E);
  // layers 1,2: 128 -> 128
  for (int l = 1; l <= 2; ++l) {
    k_gemm_wmma<<<ggrid, 128, 0, stream>>>(xbuf, Wm[l], hbuf, N, HHID, HHID);
    k_attdots<<<(N * HEADS + T - 1) / T, T, 0, stream>>>(hbuf, ASm[l], ADm[l], asb, adb, N, HEADS);
    k_aggregate<HEADS, true><<<agrid, 256, 0, stream>>>(hbuf, asb, adb, rowptr, col, ei,
                                                        Bm[l], Gm[l], BEm[l], xbuf, N, E);
  }
  // final layer: 128 -> 32, single head, no LN/ELU, write d_out
  dim3 fgrid((tilesM + 3) / 4, OUTC / 16);
  k_gemm_wmma<<<fgrid, 128, 0, stream>>>(xbuf, Wm[3], hbuf, N, HHID, OUTC);
  k_attdots<<<(N + T - 1) / T, T, 0, stream>>>(hbuf, ASm[3], ADm[3], asb, adb, N, 1);
  k_aggregate<1, false><<<agrid, 256, 0, stream>>>(hbuf, asb, adb, rowptr, col, ei,
                                                   Bm[3], nullptr, nullptr, (float*)d_out, N, E);
}